// Net_13305808683303
// MI455X (gfx1250) — compile-verified
//
#include <hip/hip_runtime.h>

typedef __attribute__((ext_vector_type(2))) float v2f;
typedef __attribute__((ext_vector_type(8))) float v8f;

#define IN_DIM 128

// ---------------- utility kernels ----------------

__global__ __launch_bounds__(256) void zero_f32(float* __restrict__ p, size_t n) {
    size_t i = (size_t)blockIdx.x * 256 + threadIdx.x;
    if (i < n) p[i] = 0.0f;
}

__global__ __launch_bounds__(256) void count_edges(const long long* __restrict__ dst,
                                                   float* __restrict__ cnt, int n_edges) {
    int e = blockIdx.x * 256 + threadIdx.x;
    if (e < n_edges) unsafeAtomicAdd(&cnt[dst[e]], 1.0f);
}

__global__ __launch_bounds__(256) void invert_cnt(float* __restrict__ cnt, int n) {
    int i = blockIdx.x * 256 + threadIdx.x;
    if (i < n) cnt[i] = 1.0f / fmaxf(cnt[i], 1.0f);
}

// One wave per edge: lane l moves feature chunk [4l, 4l+4). float4 gather from
// src row (L2-resident), 4 f32 atomic adds into dst row of agg.
__global__ __launch_bounds__(256) void sage_scatter(const float* __restrict__ feat,
                                                    const long long* __restrict__ src,
                                                    const long long* __restrict__ dst,
                                                    float* __restrict__ agg, int n_edges) {
    int e = blockIdx.x * 8 + (threadIdx.x >> 5);
    if (e >= n_edges) return;
    int lane = threadIdx.x & 31;
    long long s = src[e];
    long long d = dst[e];
    float4 v = ((const float4*)(feat + (size_t)s * IN_DIM))[lane];
    float* dp = agg + (size_t)d * IN_DIM + lane * 4;
    unsafeAtomicAdd(dp + 0, v.x);
    unsafeAtomicAdd(dp + 1, v.y);
    unsafeAtomicAdd(dp + 2, v.z);
    unsafeAtomicAdd(dp + 3, v.w);
}

// ---------------- fused SAGE GEMM ----------------
// out[m, :] = act( (agg[m,:] * cntinv[m]) @ Wl^T + b + x[m,:] @ Wr^T )
// Block = 128 threads (4 waves) handles one 16-row tile; both A operands are
// staged in LDS once (mean scaling folded into the load).  Each wave covers
// 16-column output tiles with stride 4 using V_WMMA_F32_16X16X4_F32, K-looped
// over IND in steps of 4; bias folded into the C accumulator init.
template <int IND, int OUTD, bool RELU>
__global__ __launch_bounds__(128) void sage_gemm(const float* __restrict__ agg,
                                                 const float* __restrict__ cntinv,
                                                 const float* __restrict__ xs,
                                                 const float* __restrict__ Wl,
                                                 const float* __restrict__ bias,
                                                 const float* __restrict__ Wr,
                                                 float* __restrict__ out) {
    __shared__ float lds_a[16 * IND];  // mean-aggregated neighbor features
    __shared__ float lds_x[16 * IND];  // self features
    const int tid = threadIdx.x;
    const int row_base = blockIdx.x * 16;

    // Cooperative stage: 16 x IND floats per matrix, float4 vectorized.
    const int nvec = 16 * IND / 4;
    for (int v = tid; v < nvec; v += 128) {
        int r = v / (IND / 4);
        float s = cntinv[row_base + r];
        float4 av = ((const float4*)(agg + (size_t)(row_base + r) * IND))[v % (IND / 4)];
        av.x *= s; av.y *= s; av.z *= s; av.w *= s;
        ((float4*)lds_a)[v] = av;
        ((float4*)lds_x)[v] = ((const float4*)(xs + (size_t)(row_base + r) * IND))[v % (IND / 4)];
    }
    __syncthreads();

    const int lane  = tid & 31;
    const int wave  = tid >> 5;           // 0..3
    const int n     = lane & 15;          // output column within tile / A row M
    const int khalf = (lane >> 4) * 2;    // K sub-offset per ISA f32 WMMA layout

    const float* arow = lds_a + n * IND;  // A fragment rows: M = lane & 15
    const float* xrow = lds_x + n * IND;

    const int ntiles = OUTD / 16;
    for (int t = wave; t < ntiles; t += 4) {
        // Bias depends only on output column -> broadcast into all 8 C rows.
        float bv = bias[t * 16 + n];
        v8f c;
        #pragma unroll
        for (int j = 0; j < 8; ++j) c[j] = bv;

        const float* wl_row = Wl + (size_t)(t * 16 + n) * IND;  // B[k][n] = W[n][k]
        const float* wr_row = Wr + (size_t)(t * 16 + n) * IND;

        #pragma unroll 4
        for (int kk = 0; kk < IND; kk += 4) {
            v2f a1 = { arow[kk + khalf], arow[kk + khalf + 1] };
            v2f b1 = { wl_row[kk + khalf], wl_row[kk + khalf + 1] };
            c = __builtin_amdgcn_wmma_f32_16x16x4_f32(false, a1, false, b1,
                                                      (short)0, c, false, false);
            v2f a2 = { xrow[kk + khalf], xrow[kk + khalf + 1] };
            v2f b2 = { wr_row[kk + khalf], wr_row[kk + khalf + 1] };
            c = __builtin_amdgcn_wmma_f32_16x16x4_f32(false, a2, false, b2,
                                                      (short)0, c, false, false);
        }

        // C/D layout: VGPR j holds row M = j + 8*(lane>=16), col N = lane&15.
        const int mb = (lane >> 4) * 8;
        #pragma unroll
        for (int j = 0; j < 8; ++j) {
            float vr = c[j];
            if (RELU) vr = fmaxf(vr, 0.0f);
            out[(size_t)(row_base + mb + j) * OUTD + t * 16 + n] = vr;
        }
    }
}

// ---------------- launcher ----------------

extern "C" void kernel_launch(void* const* d_in, const int* in_sizes, int n_in,
                              void* d_out, int out_size, void* d_ws, size_t ws_size,
                              hipStream_t stream) {
    const float*     x   = (const float*)d_in[0];
    const long long* ei  = (const long long*)d_in[1];  // int64 [2, E]
    const float*     W1l = (const float*)d_in[2];
    const float*     b1  = (const float*)d_in[3];
    const float*     W1r = (const float*)d_in[4];
    const float*     W2l = (const float*)d_in[5];
    const float*     b2  = (const float*)d_in[6];
    const float*     W2r = (const float*)d_in[7];
    float*           out = (float*)d_out;

    const int N = in_sizes[0] / IN_DIM;   // 100000 (divisible by 16)
    const int E = in_sizes[1] / 2;        // 1600000
    const long long* src = ei;
    const long long* dst = ei + E;

    // Workspace: [cntinv: N][agg: N*128][h: N*128]  (~103 MB)
    float* cnt = (float*)d_ws;
    float* agg = cnt + N;
    float* h   = agg + (size_t)N * IN_DIM;

    // --- degree counts (shared by both layers) + zero agg1 ---
    size_t z1 = (size_t)N * (IN_DIM + 1);
    zero_f32<<<(unsigned)((z1 + 255) / 256), 256, 0, stream>>>(cnt, z1);
    count_edges<<<(E + 255) / 256, 256, 0, stream>>>(dst, cnt, E);
    invert_cnt<<<(N + 255) / 256, 256, 0, stream>>>(cnt, N);

    // --- layer 1: scatter x -> agg, fused GEMM + bias + ReLU -> h ---
    sage_scatter<<<(E + 7) / 8, 256, 0, stream>>>(x, src, dst, agg, E);
    sage_gemm<IN_DIM, 128, true><<<N / 16, 128, 0, stream>>>(agg, cnt, x, W1l, b1, W1r, h);

    // --- layer 2: re-zero agg, scatter h -> agg, fused GEMM -> out ---
    size_t z2 = (size_t)N * IN_DIM;
    zero_f32<<<(unsigned)((z2 + 255) / 256), 256, 0, stream>>>(agg, z2);
    sage_scatter<<<(E + 7) / 8, 256, 0, stream>>>(h, src, dst, agg, E);
    sage_gemm<IN_DIM, 64, false><<<N / 16, 128, 0, stream>>>(agg, cnt, h, W2l, b2, W2r, out);
}